// ContrastiveLoss_85710367359359
// MI455X (gfx1250) — compile-verified
//
#include <hip/hip_runtime.h>
#include <math.h>

typedef float v2f __attribute__((ext_vector_type(2)));
typedef float v8f __attribute__((ext_vector_type(8)));

#define D_DIM    1024
#define EPS_COS  1e-8f
#define EP_CONST 0.001f

// One wave32 per 16-row tile. Computes e[row] = exp(cosine(q[row], im[row]))
// Dot products via V_WMMA_F32_16X16X4_F32 diagonal trick; norms via packed FMA.
__global__ __launch_bounds__(128) void cos_exp_wmma_kernel(
    const float* __restrict__ q, const float* __restrict__ im,
    float* __restrict__ e_out, int nrows)
{
    const int lane = threadIdx.x & 31;
    const int wave = threadIdx.x >> 5;
    const int tile = blockIdx.x * 4 + wave;
    const int row0 = tile << 4;
    if (row0 >= nrows) return;              // wave-uniform: EXEC all-1s for WMMA

    const int m    = lane & 15;             // row within tile
    const int half = lane >> 4;             // K split across lane halves
    const long long row = (long long)row0 + m;
    const float* qp = q  + row * (long long)D_DIM + 2 * half;
    const float* ip = im + row * (long long)D_DIM + 2 * half;

    v8f acc = {};                           // 16x16 f32 accumulator (diag = dots)
    v2f sq2 = {0.f, 0.f};                   // per-lane packed sum-of-squares (q)
    v2f si2 = {0.f, 0.f};                   // per-lane packed sum-of-squares (im)
#pragma unroll 8
    for (int k = 0; k < D_DIM; k += 4) {
        v2f aq = *(const v2f*)(qp + k);     // q  tile fragment (A layout)
        v2f ai = *(const v2f*)(ip + k);     // im tile fragment (B = im^T layout)
        acc = __builtin_amdgcn_wmma_f32_16x16x4_f32(
            /*neg_a=*/false, aq, /*neg_b=*/false, ai,
            /*c_mod=*/(short)0, acc, /*reuse_a=*/false, /*reuse_b=*/false);
        sq2 += aq * aq;                     // v_pk_fma_f32 on load pair directly
        si2 += ai * ai;
    }

    // Horizontal add of packed accumulators, then combine lane halves:
    // lane L and L+16 hold the two K-halves of row L%16.
    float sq = sq2.x + sq2.y;
    float si = si2.x + si2.y;
    float nq = sq + __shfl_xor(sq, 16, 32);
    float ni = si + __shfl_xor(si, 16, 32);

    // Diagonal extraction: diag(m) at {lane m, c[m]} for m<8, {lane m+16, c[m-8]} for m>=8.
    float v = acc[lane & 7];
    float d = __shfl(v, (m < 8) ? m : (m + 16), 32);   // lanes 0..15 hold diag(lane)

    if (lane < 16) {
        float cs = d / fmaxf(sqrtf(nq * ni), EPS_COS);
        e_out[row0 + lane] = __expf(cs);
    }
}

// Per batch item: 2 positives, 23 negatives -> (neg - pos) / (pos + neg + EP)
__global__ __launch_bounds__(256) void per_item_kernel(
    const float* __restrict__ epos, const float* __restrict__ eneg,
    float* __restrict__ per, int B)
{
    int b = blockIdx.x * blockDim.x + threadIdx.x;
    if (b >= B) return;
    float ps = epos[2 * b] + epos[2 * b + 1];
    float ns = 0.f;
    const float* p = eneg + (long long)b * 23;
#pragma unroll
    for (int i = 0; i < 23; ++i) ns += p[i];
    per[b] = (ns - ps) / (ps + ns + EP_CONST);
}

// Deterministic single-block tree reduction of B items -> scalar.
__global__ __launch_bounds__(256) void final_reduce_kernel(
    const float* __restrict__ per, float* __restrict__ out, int B)
{
    __shared__ float s[256];
    float acc = 0.f;
    for (int i = threadIdx.x; i < B; i += 256) acc += per[i];
    s[threadIdx.x] = acc;
    __syncthreads();
    for (int off = 128; off > 0; off >>= 1) {
        if ((int)threadIdx.x < off) s[threadIdx.x] += s[threadIdx.x + off];
        __syncthreads();
    }
    if (threadIdx.x == 0) out[0] = s[0];
}

extern "C" void kernel_launch(void* const* d_in, const int* in_sizes, int n_in,
                              void* d_out, int out_size, void* d_ws, size_t ws_size,
                              hipStream_t stream) {
    const float* q_pos  = (const float*)d_in[0];   // [2B, 1024]
    const float* q_neg  = (const float*)d_in[1];   // [23B, 1024]
    const float* im_pos = (const float*)d_in[2];   // [2B, 1024]
    const float* im_neg = (const float*)d_in[3];   // [23B, 1024]

    const int npos = in_sizes[0] / D_DIM;          // 2B  = 8192
    const int nneg = in_sizes[1] / D_DIM;          // 23B = 94208
    const int B    = npos / 2;                     // 4096

    // workspace layout: epos[npos] | eneg[nneg] | per[B]  (~416 KB)
    float* epos = (float*)d_ws;
    float* eneg = epos + npos;
    float* per  = eneg + nneg;

    {
        int tiles  = (npos + 15) / 16;             // 512
        int blocks = (tiles + 3) / 4;
        cos_exp_wmma_kernel<<<blocks, 128, 0, stream>>>(q_pos, im_pos, epos, npos);
    }
    {
        int tiles  = (nneg + 15) / 16;             // 5888
        int blocks = (tiles + 3) / 4;
        cos_exp_wmma_kernel<<<blocks, 128, 0, stream>>>(q_neg, im_neg, eneg, nneg);
    }
    per_item_kernel<<<(B + 255) / 256, 256, 0, stream>>>(epos, eneg, per, B);
    final_reduce_kernel<<<1, 256, 0, stream>>>(per, (float*)d_out, B);
}